// SeastarGCNLayer_14181982011586
// MI455X (gfx1250) — compile-verified
//
#include <hip/hip_runtime.h>

// SeastarGCNLayer for MI455X (gfx1250, wave32).
//   Phase 0: zero accumulator; transpose W (128 KB) for contiguous B fragments
//   Phase 1: hw = h @ W   -- fp32 WMMA, one wave per 16x128 row-block
//                            (8 accumulators, A reused 8x per K-step)
//   Phase 2: edge gather/scatter -- wave-per-edge, b128 loads + f32 atomics
//   Phase 3: out = relu(agg*norm + bias)

typedef __attribute__((ext_vector_type(2))) float v2f;
typedef __attribute__((ext_vector_type(8))) float v8f;

#define IN_FEATS  256
#define OUT_FEATS 128

// ---------------------------------------------------------------- zero d_out
__global__ void gcn_zero_kernel(float* __restrict__ out, int n4) {
    int i = blockIdx.x * blockDim.x + threadIdx.x;
    if (i < n4) {
        ((float4*)out)[i] = make_float4(0.f, 0.f, 0.f, 0.f);
    }
}

// ------------------------------------------------ transpose W: wt[n,k] = w[k,n]
__global__ void gcn_transpose_w_kernel(const float* __restrict__ w,
                                       float* __restrict__ wt) {
    int i = blockIdx.x * blockDim.x + threadIdx.x;   // over 256*128
    if (i < IN_FEATS * OUT_FEATS) {
        int k = i / OUT_FEATS;
        int n = i % OUT_FEATS;
        wt[(size_t)n * IN_FEATS + k] = w[i];
    }
}

// ------------------------------------------------- Phase 1: fp32 WMMA GEMM
// One wave per 16-row M-tile; computes the full 16x128 output block with
// 8 WMMA accumulators, reusing the A fragment across all 8 N-tiles.
__global__ __launch_bounds__(128) void gcn_gemm_wmma_kernel(
        const float* __restrict__ h,     // [n_nodes, 256]
        const float* __restrict__ wt,    // [128, 256]  (W transposed)
        float* __restrict__ hw,          // [n_nodes, 128]
        int n_nodes) {
    const int lane = threadIdx.x & 31;
    const int wave = threadIdx.x >> 5;
    const int mt   = blockIdx.x * 4 + wave;
    if (mt * 16 >= n_nodes) return;                  // wave-uniform branch

    // A (16x4 f32): lanes 0-15 -> M, K=kb..kb+1 ; lanes 16-31 -> M, K=kb+2..3
    const int m  = lane & 15;
    const int kb = (lane >> 4) << 1;                 // 0 or 2
    const float* __restrict__ arow = h + (size_t)(mt * 16 + m) * IN_FEATS + kb;

    // B (4x16 f32): lane group picks K pair; element (K, N) lives at wt[N*256+K]
    // -> per-lane fragment is contiguous in K: one global_load_b64 per tile.
    const int n = lane & 15;
    const float* __restrict__ brow = wt + (size_t)n * IN_FEATS + kb;

    v8f c[8] = {};
#pragma unroll 4
    for (int k = 0; k < IN_FEATS; k += 4) {
        v2f a = *(const v2f*)(arow + k);             // global_load_b64
#pragma unroll
        for (int t = 0; t < 8; ++t) {
            v2f b = *(const v2f*)(brow + (size_t)(t * 16) * IN_FEATS + k);
            // 8 args: (neg_a, A, neg_b, B, c_mod, C, reuse_a, reuse_b)
            c[t] = __builtin_amdgcn_wmma_f32_16x16x4_f32(
                       false, a, false, b, (short)0, c[t], false, false);
        }
    }

    // C/D layout: lane = 16*(m>=8) + n, VGPR r holds row m = 8*(lane>>4) + r
    const int mb = (lane >> 4) << 3;                 // 0 or 8
    float* __restrict__ obase =
        hw + (size_t)(mt * 16 + mb) * OUT_FEATS + n;
#pragma unroll
    for (int t = 0; t < 8; ++t) {
#pragma unroll
        for (int r = 0; r < 8; ++r) {
            obase[(size_t)r * OUT_FEATS + t * 16] = c[t][r];
        }
    }
}

// ------------------------------------- Phase 2: per-edge gather + atomic scatter
// One wave32 per edge: lane l handles feats [4l, 4l+4).
__global__ __launch_bounds__(256) void gcn_edge_scatter_kernel(
        const float* __restrict__ hw,          // [n_nodes, 128]
        const float* __restrict__ norm,        // [n_nodes]
        const float* __restrict__ edge_weight, // [n_edges]
        const int*   __restrict__ edge_src,    // [n_edges]
        const int*   __restrict__ edge_dst,    // [n_edges]
        float* __restrict__ agg,               // [n_nodes, 128] (accumulator)
        int n_edges) {
    const int lane = threadIdx.x & 31;
    const int e    = (blockIdx.x * blockDim.x + threadIdx.x) >> 5;
    if (e >= n_edges) return;                  // wave-uniform branch

    const int   s  = edge_src[e];
    const int   d  = edge_dst[e];
    const float sc = norm[s] * edge_weight[e];

    const float4 v = ((const float4*)(hw + (size_t)s * OUT_FEATS))[lane];
    float* __restrict__ ad = agg + (size_t)d * OUT_FEATS + lane * 4;
    atomicAdd(ad + 0, sc * v.x);
    atomicAdd(ad + 1, sc * v.y);
    atomicAdd(ad + 2, sc * v.z);
    atomicAdd(ad + 3, sc * v.w);
}

// ------------------------------------------- Phase 3: scale + bias + relu
__global__ void gcn_finalize_kernel(
        float* __restrict__ out,              // [n_nodes, 128] in/out
        const float* __restrict__ norm,       // [n_nodes]
        const float* __restrict__ bias,       // [128]
        int n_nodes) {
    const int idx   = blockIdx.x * blockDim.x + threadIdx.x;  // float4 index
    const int total = n_nodes * (OUT_FEATS / 4);
    if (idx >= total) return;
    const int node = idx >> 5;                // / (128/4)
    const int f4   = idx & 31;

    const float  nn = norm[node];
    const float4 b  = ((const float4*)bias)[f4];
    float4 a = ((float4*)out)[idx];
    a.x = fmaxf(a.x * nn + b.x, 0.f);
    a.y = fmaxf(a.y * nn + b.y, 0.f);
    a.z = fmaxf(a.z * nn + b.z, 0.f);
    a.w = fmaxf(a.w * nn + b.w, 0.f);
    ((float4*)out)[idx] = a;
}

extern "C" void kernel_launch(void* const* d_in, const int* in_sizes, int n_in,
                              void* d_out, int out_size, void* d_ws, size_t ws_size,
                              hipStream_t stream) {
    // setup_inputs order: h, weight, bias, norm, edge_weight, edge_src, edge_dst
    const float* h           = (const float*)d_in[0];
    const float* weight      = (const float*)d_in[1];
    const float* bias        = (const float*)d_in[2];
    const float* norm        = (const float*)d_in[3];
    const float* edge_weight = (const float*)d_in[4];
    const int*   edge_src    = (const int*)d_in[5];
    const int*   edge_dst    = (const int*)d_in[6];

    const int n_nodes = in_sizes[3];          // 50000
    const int n_edges = in_sizes[4];          // 800000

    // workspace layout: hw [n_nodes*128 f32] | wt [128*256 f32]
    float* hw  = (float*)d_ws;
    float* wt  = hw + (size_t)n_nodes * OUT_FEATS;
    float* agg = (float*)d_out;               // accumulate directly into d_out

    // 0a) zero accumulator (every call: graph replays reuse d_out)
    {
        const int n4 = n_nodes * (OUT_FEATS / 4);
        gcn_zero_kernel<<<(n4 + 255) / 256, 256, 0, stream>>>(agg, n4);
    }
    // 0b) transpose W so B fragments are contiguous b64 loads
    {
        const int nelem = IN_FEATS * OUT_FEATS;
        gcn_transpose_w_kernel<<<(nelem + 255) / 256, 256, 0, stream>>>(weight, wt);
    }

    // 1) WMMA GEMM: one wave per 16-row block, 4 waves per 128-thread block
    {
        const int mtiles = (n_nodes + 15) / 16;       // 3125
        const int blocks = (mtiles + 3) / 4;
        gcn_gemm_wmma_kernel<<<blocks, 128, 0, stream>>>(h, wt, hw, n_nodes);
    }

    // 2) edge scatter: one wave per edge, 8 edges per 256-thread block
    {
        const int blocks = (n_edges + 7) / 8;
        gcn_edge_scatter_kernel<<<blocks, 256, 0, stream>>>(
            hw, norm, edge_weight, edge_src, edge_dst, agg, n_edges);
    }

    // 3) finalize: relu(agg*norm + bias), in place on d_out
    {
        const int total = n_nodes * (OUT_FEATS / 4);
        gcn_finalize_kernel<<<(total + 255) / 256, 256, 0, stream>>>(
            (float*)d_out, norm, bias, n_nodes);
    }
}